// ClsSO3VoxConvModel_62337155334545
// MI455X (gfx1250) — compile-verified
//
#include <hip/hip_runtime.h>
#include <math.h>

// ---------------------------------------------------------------------------
// CDNA5 (gfx1250) SO(3) voxel CNN on V_WMMA_F32_16X16X4_F32 (full fp32).
// Weights are pre-packed per conv into a pair-interleaved blocked layout
// [mt][kPair][lane][2] with each cin-run padded to a multiple of 4 taps and
// zero-filled: A fragments become single aligned global_load_b64 at immediate
// offsets, and all edge-clamp logic vanishes (padded taps are true zeros).
// ---------------------------------------------------------------------------

typedef float v2f __attribute__((ext_vector_type(2)));
typedef float v8f __attribute__((ext_vector_type(8)));

#define WMMA_F32(a, b, c) \
    __builtin_amdgcn_wmma_f32_16x16x4_f32(false, (a), false, (b), (short)0, (c), false, false)

__device__ __forceinline__ float elu1(float v) {
    return v > 0.f ? v : (expf(v) - 1.f);
}

__device__ __forceinline__ float maskf(float v, bool keep) {
    return __uint_as_float(__float_as_uint(v) & (keep ? 0xFFFFFFFFu : 0u));
}

// ---------------------------------------------------------------------------
// Pack W[M][K] (K = nRuns*Krun) into Wb[mt][p][lm][2]:
//   g = run*RUN + tap (RUN = Krun padded to mult of 4), p = g/2, j = g&1
//   value = (m<M && tap<Krun) ? W[m*K + run*Krun + tap] : 0
// ---------------------------------------------------------------------------
__global__ __launch_bounds__(256) void pack_weights(
    const float* __restrict__ W, float* __restrict__ Wb,
    int M, int K, int Krun, int RUN, int nRuns)
{
    const int mt = blockIdx.y;
    const long long Gtot = (long long)nRuns * RUN;
    const long long i = (long long)blockIdx.x * 256 + threadIdx.x;  // over Gtot*16
    if (i >= Gtot * 16) return;
    const int g = (int)(i >> 4), lm = (int)(i & 15);
    const int run = g / RUN, tap = g - run * RUN;
    const int m = mt * 16 + lm;
    const bool ok = (m < M) && (tap < Krun);
    const float v = ok ? W[(long long)m * K + run * Krun + tap] : 0.f;
    const int p = g >> 1, j = g & 1;
    Wb[((long long)mt * Gtot + p * 2) * 16 + lm * 2 + j] = v;
}

// ---------------------------------------------------------------------------
// Implicit-GEMM conv3d: M = Cout, N = 4*outD^3, K = Cin*KK^3.
// One wave owns a 64(M) x 16(N) tile; one gathered B fragment feeds 4 WMMAs.
// WMMA f32 16x16x4 layout: A/B lane->(M|N)=lane&15, vgpr v -> K=v+2*(lane>>4);
// C lane->N=lane&15, vgpr v -> M=v+8*(lane>>4).
// ---------------------------------------------------------------------------
template <int KK>
__global__ __launch_bounds__(256) void conv3d_igemm_wmma(
    const float* __restrict__ X, const float* __restrict__ Wb, float* __restrict__ Y,
    int Cin, int Cout, int inD, int outD, int pad, int stride)
{
    constexpr int KK2 = KK * KK;
    constexpr int KK3 = KK2 * KK;
    constexpr int RUN = (KK3 + 3) & ~3;               // padded taps per cin
    constexpr int NSTEP = RUN / 4;
    const int lane = threadIdx.x & 31;
    const int wid  = threadIdx.x >> 5;
    const int half = lane >> 4;
    const int lm   = lane & 15;
    const int kBase = 2 * half;

    const int K = Cin * KK3;
    const int M = Cout;
    const int nvoxOut = outD * outD * outD;
    const int N = 4 * nvoxOut;
    const int mSuper = (M + 63) >> 6;
    const int nTiles = (N + 15) >> 4;
    const int tile = blockIdx.x * 8 + wid;            // wave-uniform
    if (tile >= mSuper * nTiles) return;
    const int ms = (tile % mSuper) * 64;
    const int nt = tile / mSuper;

    const int n = nt * 16 + lm;
    const bool nOK = (n < N);
    int t = nOK ? n : 0;                              // fake cols -> col 0, not stored
    const int ow = t % outD; t /= outD;
    const int oh = t % outD; t /= outD;
    const int od = t % outD; t /= outD;
    const int bN = t;

    const int id0 = od * stride - pad;
    const int ih0 = oh * stride - pad;
    const int iw0 = ow * stride - pad;
    const int inSq = inD * inD;
    const int inSl = inSq * inD;
    const float* xc = X + (long long)bN * Cin * inSl;

    const v8f vz = {0.f, 0.f, 0.f, 0.f, 0.f, 0.f, 0.f, 0.f};
    v8f acc[4];
#pragma unroll
    for (int tt = 0; tt < 4; ++tt) acc[tt] = vz;

    if constexpr (KK == 1) {
        // 1x1x1 conv (pad=0, stride=1): K = Cin padded to RUN1 (mult of 4).
        const int RUN1 = (K + 3) & ~3;
        const float* pa[4];
#pragma unroll
        for (int tt = 0; tt < 4; ++tt)
            pa[tt] = Wb + ((long long)((ms >> 4) + tt) * RUN1 + kBase) * 16 + lm * 2;

        const int b0 = (od * inD + oh) * inD + ow;
        const float* pX = xc + b0 + (long long)kBase * inSl;
        const int K4 = K & ~3;
        int k0 = 0;
        for (; k0 < K4; k0 += 4) {
            v2f bf;
            bf[0] = pX[0];
            bf[1] = pX[inSl];
#pragma unroll
            for (int tt = 0; tt < 4; ++tt) {
                const v2f af = *(const v2f*)(pa[tt]);
                acc[tt] = WMMA_F32(af, bf, acc[tt]);
            }
            pX += 4LL * inSl;
#pragma unroll
            for (int tt = 0; tt < 4; ++tt) pa[tt] += 64;
        }
        if (k0 < K) {                                 // masked tail (1..3 cins)
            const float* pX0 = xc + b0;
            v2f bf;
#pragma unroll
            for (int v = 0; v < 2; ++v) {
                int cin = k0 + v + kBase;
                const bool ok = (cin < K);
                cin = ok ? cin : K - 1;
                bf[v] = maskf(pX0[(long long)cin * inSl], ok);
            }
#pragma unroll
            for (int tt = 0; tt < 4; ++tt) {
                const v2f af = *(const v2f*)(pa[tt]);  // padded zeros past K
                acc[tt] = WMMA_F32(af, bf, acc[tt]);
            }
        }
    } else if constexpr (KK == 3) {
        const long long Gtot = (long long)Cin * RUN;
        const float* pa[4];
#pragma unroll
        for (int tt = 0; tt < 4; ++tt)
            pa[tt] = Wb + ((long long)((ms >> 4) + tt) * Gtot + kBase) * 16 + lm * 2;

        // Hoist the cin-invariant gather pattern: 14 offsets + 14 mask words.
        int      goff[2 * NSTEP];
        unsigned gmw [2 * NSTEP];
#pragma unroll
        for (int s = 0; s < NSTEP; ++s)
#pragma unroll
            for (int v = 0; v < 2; ++v) {
                const int tap = s * 4 + v + kBase;
                const int kd = tap / KK2;
                const int kh = (tap - kd * KK2) / KK;
                const int kw = tap - kd * KK2 - kh * KK;
                const int id = id0 + kd, ih = ih0 + kh, iw = iw0 + kw;
                const bool ok = (tap < KK3) &&
                    (unsigned)id < (unsigned)inD &&
                    (unsigned)ih < (unsigned)inD &&
                    (unsigned)iw < (unsigned)inD;
                goff[s * 2 + v] = ok ? (id * inSq + ih * inD + iw) : 0;
                gmw [s * 2 + v] = ok ? 0xFFFFFFFFu : 0u;
            }

        for (int cin = 0; cin < Cin; ++cin) {
#pragma unroll
            for (int s = 0; s < NSTEP; ++s) {
                v2f bf;
                bf[0] = __uint_as_float(
                    __float_as_uint(xc[goff[s * 2 + 0]]) & gmw[s * 2 + 0]);
                bf[1] = __uint_as_float(
                    __float_as_uint(xc[goff[s * 2 + 1]]) & gmw[s * 2 + 1]);
#pragma unroll
                for (int tt = 0; tt < 4; ++tt) {
                    const v2f af = *(const v2f*)(pa[tt] + s * 64);
                    acc[tt] = WMMA_F32(af, bf, acc[tt]);
                }
            }
            xc += inSl;
#pragma unroll
            for (int tt = 0; tt < 4; ++tt) pa[tt] += RUN * 16;
        }
    } else {
        // KK == 5 (stem, Cin == 1): per-tap compute, runs once per tap.
        const long long Gtot = (long long)Cin * RUN;
        const float* pa[4];
#pragma unroll
        for (int tt = 0; tt < 4; ++tt)
            pa[tt] = Wb + ((long long)((ms >> 4) + tt) * Gtot + kBase) * 16 + lm * 2;

        for (int cin = 0; cin < Cin; ++cin) {
#pragma unroll
            for (int s = 0; s < NSTEP; ++s) {
                v2f bf;
#pragma unroll
                for (int v = 0; v < 2; ++v) {
                    const int tap = s * 4 + v + kBase;
                    const int kd = tap / KK2;
                    const int kh = (tap - kd * KK2) / KK;
                    const int kw = tap - kd * KK2 - kh * KK;
                    const int id = id0 + kd, ih = ih0 + kh, iw = iw0 + kw;
                    const bool ok = (tap < KK3) &&
                        (unsigned)id < (unsigned)inD &&
                        (unsigned)ih < (unsigned)inD &&
                        (unsigned)iw < (unsigned)inD;
                    const int off = ok ? (id * inSq + ih * inD + iw) : 0;
                    bf[v] = maskf(xc[off], ok);
                }
#pragma unroll
                for (int tt = 0; tt < 4; ++tt) {
                    const v2f af = *(const v2f*)(pa[tt] + s * 64);
                    acc[tt] = WMMA_F32(af, bf, acc[tt]);
                }
            }
            xc += inSl;
#pragma unroll
            for (int tt = 0; tt < 4; ++tt) pa[tt] += RUN * 16;
        }
    }

    if (nOK) {
        float* yb = Y + (long long)bN * Cout * nvoxOut +
                    ((long long)od * outD + oh) * outD + ow;
#pragma unroll
        for (int tt = 0; tt < 4; ++tt)
#pragma unroll
            for (int v = 0; v < 8; ++v) {
                const int row = ms + tt * 16 + v + 8 * half;
                if (row < M) yb[(long long)row * nvoxOut] = acc[tt][v];
            }
    }
}

// ---------------------------------------------------------------------------
// Fused per-voxel Fourier ELU, templated on NVOX and F so every global access
// is base-pointer + compile-time immediate offset (d*NVOX*4 < 8MB IOFFSET).
// Padded sA column 35 is zero (annihilates the d=35 lane); fake columns are
// never stored; the <= 35*NVOX overread stays inside the MID ws region.
// ---------------------------------------------------------------------------
#define LDSROW 18
template <int NVOX, int F>
__global__ __launch_bounds__(256) void fourier_elu_wmma(
    float* __restrict__ Y, const float* __restrict__ A, const float* __restrict__ Ai)
{
    __shared__ float sA[64 * 36];
    __shared__ float sAi[48 * 60];
    __shared__ float sG[8 * 64 * LDSROW];

    for (int i = threadIdx.x; i < 64 * 36; i += 256) {
        const int r = i / 36, c = i - r * 36;
        sA[i] = (r < 60 && c < 35) ? A[r * 35 + c] : 0.f;
    }
    for (int i = threadIdx.x; i < 48 * 60; i += 256)
        sAi[i] = (i < 35 * 60) ? Ai[i] : 0.f;      // same row stride (60)
    __syncthreads();

    const int lane = threadIdx.x & 31;
    const int wid  = threadIdx.x >> 5;
    const int half = lane >> 4;
    const int lm   = lane & 15;
    const int kBase = 2 * half;
    constexpr int C  = F * 35;
    constexpr int NT = 4 * F * NVOX;

    const int nTiles = (NT + 15) >> 4;
    const int tile = blockIdx.x * 8 + wid;
    if (tile >= nTiles) return;

    int n = tile * 16 + lm;
    const bool nOK = (n < NT);
    n = nOK ? n : 0;
    const int vox = n % NVOX;
    const int ff  = (n / NVOX) % F;
    const int bN  = n / (NVOX * F);
    const long long base = ((long long)bN * C + ff * 35) * (long long)NVOX + vox;
    const float* pB = Y + base + kBase * NVOX;

    const v8f vz = {0.f, 0.f, 0.f, 0.f, 0.f, 0.f, 0.f, 0.f};

    // forward: G(64x16) = Apad(64x36) * Xpad(36x16)
    v8f acc[4];
#pragma unroll
    for (int tt = 0; tt < 4; ++tt) acc[tt] = vz;

#pragma unroll
    for (int k0 = 0; k0 < 36; k0 += 4) {
        v2f bf;
        bf[0] = pB[k0 * NVOX];
        bf[1] = pB[(k0 + 1) * NVOX];
#pragma unroll
        for (int tt = 0; tt < 4; ++tt) {
            v2f af;
            af[0] = sA[(tt * 16 + lm) * 36 + k0 + kBase];
            af[1] = sA[(tt * 16 + lm) * 36 + k0 + kBase + 1];
            acc[tt] = WMMA_F32(af, bf, acc[tt]);
        }
    }

    // ELU + stage to LDS as B matrix of the inverse GEMM
    float* G = &sG[wid * 64 * LDSROW];
#pragma unroll
    for (int tt = 0; tt < 4; ++tt)
#pragma unroll
        for (int v = 0; v < 8; ++v)
            G[(tt * 16 + v + 8 * half) * LDSROW + lm] = elu1(acc[tt][v]);

    // inverse: X'(48x16) = Aipad(48x60) * G(60x16), rows 0..34 valid
    v8f acc2[3];
#pragma unroll
    for (int tt = 0; tt < 3; ++tt) acc2[tt] = vz;

#pragma unroll
    for (int k0 = 0; k0 < 60; k0 += 4) {
        v2f bf;
        bf[0] = G[(k0 + kBase) * LDSROW + lm];
        bf[1] = G[(k0 + kBase + 1) * LDSROW + lm];
#pragma unroll
        for (int tt = 0; tt < 3; ++tt) {
            v2f af;
            af[0] = sAi[(tt * 16 + lm) * 60 + k0 + kBase];
            af[1] = sAi[(tt * 16 + lm) * 60 + k0 + kBase + 1];
            acc2[tt] = WMMA_F32(af, bf, acc2[tt]);
        }
    }

    if (nOK) {
        float* pS = Y + base + (long long)(8 * half) * NVOX;
#pragma unroll
        for (int tt = 0; tt < 3; ++tt)
#pragma unroll
            for (int v = 0; v < 8; ++v) {
                if (tt * 16 + v + 8 * half < 35)    // folds for tt<2
                    pS[(tt * 16 + v) * NVOX] = acc2[tt][v];
            }
    }
}

// ---------------------------------------------------------------------------
// IIDBatchNorm stats: one block per field, division-free pointer-bump loops,
// double accumulators, deterministic fixed-tree reduction (graph-replay safe).
// ---------------------------------------------------------------------------
__global__ __launch_bounds__(256) void bn_stats(
    const float* __restrict__ Y, float* __restrict__ stats, int f, int nvox)
{
    __shared__ double red[512];
    const int ff = blockIdx.x;
    const int C = f * 35;
    double s0 = 0.0, s2 = 0.0;
    for (int bN = 0; bN < 4; ++bN) {
        const float* pb = Y + ((long long)bN * C + (long long)ff * 35) * nvox;
        for (int vox = threadIdx.x; vox < nvox; vox += 256) {   // d == 0
            const float v = pb[vox];
            s0 += (double)v;
            s2 += (double)v * (double)v;
        }
        for (int d = 1; d < 35; ++d) {
            const float* p = pb + (long long)d * nvox;
            for (int vox = threadIdx.x; vox < nvox; vox += 256) {
                const float v = p[vox];
                s2 += (double)v * (double)v;
            }
        }
    }
    red[threadIdx.x] = s0;
    red[256 + threadIdx.x] = s2;
    __syncthreads();
    for (int s = 128; s > 0; s >>= 1) {
        if ((int)threadIdx.x < s) {
            red[threadIdx.x] += red[threadIdx.x + s];
            red[256 + threadIdx.x] += red[256 + threadIdx.x + s];
        }
        __syncthreads();
    }
    if (threadIdx.x == 0) {
        stats[2 * ff] = (float)red[0];
        stats[2 * ff + 1] = (float)red[256];
    }
}

// One block per (batch, channel): channel decode once, then a pure stream.
__global__ __launch_bounds__(256) void bn_apply(
    float* __restrict__ Y, const float* __restrict__ stats,
    const float* __restrict__ w, const float* __restrict__ bias,
    int C, int nvox)
{
    const int bc = blockIdx.x;                 // b*C + c
    const int c = bc % C;
    const int ff = c / 35;
    const int d = c - ff * 35;
    const float cnt = 4.f * (float)nvox;
    const float s0 = stats[2 * ff], s2 = stats[2 * ff + 1];
    const float m = s0 / cnt;
    const float var = (s2 - s0 * s0 / cnt) / (cnt * 35.f);
    const float sc = w[ff] / sqrtf(var + 1e-5f);
    float* p = Y + (long long)bc * nvox;
    if (d == 0) {
        const float bb = bias[ff];
        for (int vox = threadIdx.x; vox < nvox; vox += 256)
            p[vox] = (p[vox] - m) * sc + bb;
    } else {
        for (int vox = threadIdx.x; vox < nvox; vox += 256)
            p[vox] *= sc;
    }
}

// AvgPool3d(k=2,s=2,pad=1,count_include_pad) of S added into Y (residual).
__global__ __launch_bounds__(256) void avgpool2_add(
    const float* __restrict__ S, float* __restrict__ Y, int inD, int outD)
{
    const int bc = blockIdx.x;
    const int oVol = outD * outD * outD;
    const float* sp = S + (long long)bc * inD * inD * inD;
    float* yp = Y + (long long)bc * oVol;
    for (int i = threadIdx.x; i < oVol; i += 256) {
        int t = i;
        const int ow = t % outD; t /= outD;
        const int oh = t % outD; t /= outD;
        const int od = t;
        float s = 0.f;
#pragma unroll
        for (int dz = 0; dz < 2; ++dz)
#pragma unroll
            for (int dy = 0; dy < 2; ++dy)
#pragma unroll
                for (int dx = 0; dx < 2; ++dx) {
                    const int id = od * 2 - 1 + dz;
                    const int ih = oh * 2 - 1 + dy;
                    const int iw = ow * 2 - 1 + dx;
                    if ((unsigned)id < (unsigned)inD && (unsigned)ih < (unsigned)inD &&
                        (unsigned)iw < (unsigned)inD)
                        s += sp[((long long)id * inD + ih) * inD + iw];
                }
        yp[i] += s * 0.125f;
    }
}

// norm_pool: [4,4480] -> [4,128] field norms
__global__ void norm_pool_k(const float* __restrict__ X, float* __restrict__ O)
{
    const int i = blockIdx.x * 256 + threadIdx.x;
    if (i >= 4 * 128) return;
    const int bN = i >> 7, j = i & 127;
    float s = 1e-12f;
#pragma unroll
    for (int d = 0; d < 35; ++d) {
        const float v = X[bN * 4480 + j * 35 + d];
        s += v * v;
    }
    O[i] = sqrtf(s);
}

// Whole FC head in one block: fc1 -> BN(batch) -> elu -> fc2 -> BN -> elu -> fc3
__global__ __launch_bounds__(128) void fc_head(
    const float* __restrict__ hin,
    const float* __restrict__ fc1w, const float* __restrict__ fc1b,
    const float* __restrict__ fc1gw, const float* __restrict__ fc1gb,
    const float* __restrict__ fc2w, const float* __restrict__ fc2b,
    const float* __restrict__ fc2gw, const float* __restrict__ fc2gb,
    const float* __restrict__ fc3w, const float* __restrict__ fc3b,
    float* __restrict__ out)
{
    __shared__ float h[4][128];
    const int j = threadIdx.x;
#pragma unroll
    for (int b = 0; b < 4; ++b) h[b][j] = hin[b * 128 + j];
    __syncthreads();

    for (int layer = 0; layer < 2; ++layer) {
        const float* W  = layer ? fc2w  : fc1w;
        const float* Bv = layer ? fc2b  : fc1b;
        const float* Gw = layer ? fc2gw : fc1gw;
        const float* Gb = layer ? fc2gb : fc1gb;
        float y[4];
#pragma unroll
        for (int b = 0; b < 4; ++b) {
            float s = Bv[j];
            for (int k = 0; k < 128; ++k) s += W[j * 128 + k] * h[b][k];
            y[b] = s;
        }
        const float m = 0.25f * (y[0] + y[1] + y[2] + y[3]);
        float var = 0.f;
#pragma unroll
        for (int b = 0; b < 4; ++b) { const float d = y[b] - m; var += d * d; }
        var *= 0.25f;
        const float sc = 1.f / sqrtf(var + 1e-5f);
        __syncthreads();
#pragma unroll
        for (int b = 0; b < 4; ++b)
            h[b][j] = elu1((y[b] - m) * sc * Gw[j] + Gb[j]);
        __syncthreads();
    }

    if (j < 40) {
#pragma unroll
        for (int b = 0; b < 4; ++b) {
            float s = fc3b[j];
            for (int k = 0; k < 128; ++k) s += fc3w[j * 128 + k] * h[b][k];
            out[b * 40 + j] = s;
        }
    }
}

// ---------------------------------------------------------------------------
// Host orchestration. Workspace layout (floats), peak ~377 MB:
//   P0 [6.0M] | P1 [2.0M] | MID [35.3M] | SB [11.3M] | WPK [39.5M packed W]
//   STAT [64] | W5O [17,920] | FCH [512]
// ---------------------------------------------------------------------------
extern "C" void kernel_launch(void* const* d_in, const int* in_sizes, int n_in,
                              void* d_out, int out_size, void* d_ws, size_t ws_size,
                              hipStream_t stream)
{
    (void)in_sizes; (void)n_in; (void)out_size; (void)ws_size;

    const float* x   = (const float*)d_in[0];
    const float* w0  = (const float*)d_in[1];
    const float* bw1[4] = {(const float*)d_in[2],  (const float*)d_in[9],
                           (const float*)d_in[16], (const float*)d_in[23]};
    const float* bnw[4] = {(const float*)d_in[3],  (const float*)d_in[10],
                           (const float*)d_in[17], (const float*)d_in[24]};
    const float* bnb[4] = {(const float*)d_in[4],  (const float*)d_in[11],
                           (const float*)d_in[18], (const float*)d_in[25]};
    const float* bA [4] = {(const float*)d_in[5],  (const float*)d_in[12],
                           (const float*)d_in[19], (const float*)d_in[26]};
    const float* bAi[4] = {(const float*)d_in[6],  (const float*)d_in[13],
                           (const float*)d_in[20], (const float*)d_in[27]};
    const float* bw2[4] = {(const float*)d_in[7],  (const float*)d_in[14],
                           (const float*)d_in[21], (const float*)d_in[28]};
    const float* bsw[4] = {(const float*)d_in[8],  (const float*)d_in[15],
                           (const float*)d_in[22], (const float*)d_in[29]};
    const float* w5    = (const float*)d_in[30];
    const float* fc1w  = (const float*)d_in[31];
    const float* fc1b  = (const float*)d_in[32];
    const float* fc1gw = (const float*)d_in[33];
    const float* fc1gb = (const float*)d_in[34];
    const float* fc2w  = (const float*)d_in[35];
    const float* fc2b  = (const float*)d_in[36];
    const float* fc2gw = (const float*)d_in[37];
    const float* fc2gb = (const float*)d_in[38];
    const float* fc3w  = (const float*)d_in[39];
    const float* fc3b  = (const float*)d_in[40];

    float* ws   = (float*)d_ws;
    float* P0   = ws;                    // 6,000,000
    float* P1   = ws + 6000000LL;        // 2,000,000
    float* MID  = ws + 8000000LL;        // 35,300,000
    float* SB   = ws + 43300000LL;       // 11,300,000
    float* WPK  = ws + 54600000LL;       // 39,500,000 (packed weights, reused)
    float* STAT = ws + 94100000LL;       // 64
    float* W5O  = ws + 94100064LL;       // 17,920
    float* FCH  = ws + 94118000LL;       // 512

    // pack W into the pair-interleaved padded layout, then launch the conv
    auto conv = [&](int kk, const float* Xp, const float* Wp, float* Yp,
                    int Cin, int Cout, int inD, int outD, int pad, int stride) {
        const int kk3 = kk * kk * kk;
        const int K = Cin * kk3;
        const int Krun  = (kk == 1) ? K : kk3;
        const int RUN   = (Krun + 3) & ~3;
        const int nRuns = (kk == 1) ? 1 : Cin;
        const int mT = ((Cout + 63) / 64) * 4;
        const long long Gtot = (long long)nRuns * RUN;
        dim3 pg((unsigned)((Gtot * 16 + 255) / 256), (unsigned)mT);
        pack_weights<<<pg, 256, 0, stream>>>(Wp, WPK, Cout, K, Krun, RUN, nRuns);

        const long long N = 4LL * outD * outD * outD;
        const long long tiles = (long long)((Cout + 63) / 64) * ((N + 15) / 16);
        const int blocks = (int)((tiles + 7) / 8);
        switch (kk) {
            case 1:
                conv3d_igemm_wmma<1><<<blocks, 256, 0, stream>>>(
                    Xp, WPK, Yp, Cin, Cout, inD, outD, pad, stride);
                break;
            case 3:
                conv3d_igemm_wmma<3><<<blocks, 256, 0, stream>>>(
                    Xp, WPK, Yp, Cin, Cout, inD, outD, pad, stride);
                break;
            default:
                conv3d_igemm_wmma<5><<<blocks, 256, 0, stream>>>(
                    Xp, WPK, Yp, Cin, Cout, inD, outD, pad, stride);
                break;
        }
    };

    auto fourier = [&](int nvox, int f, const float* Ap, const float* Aip) {
        const long long Ntot = 4LL * f * nvox;
        const int blocks = (int)(((Ntot + 15) / 16 + 7) / 8);
        if (nvox == 35937)
            fourier_elu_wmma<35937, 7><<<blocks, 256, 0, stream>>>(MID, Ap, Aip);
        else if (nvox == 4913)
            fourier_elu_wmma<4913, 14><<<blocks, 256, 0, stream>>>(MID, Ap, Aip);
        else if (nvox == 729)
            fourier_elu_wmma<729, 14><<<blocks, 256, 0, stream>>>(MID, Ap, Aip);
        else
            fourier_elu_wmma<125, 27><<<blocks, 256, 0, stream>>>(MID, Ap, Aip);
    };

    // stem: [4,1,33^3] -> [4,39,33^3]
    conv(5, x, w0, P0, 1, 39, 33, 33, 2, 1);

    struct Blk { int Cin, Cmid, f, Cout, inD, outD; };
    const Blk blk[4] = {{39, 245, 7, 78, 33, 17},
                        {78, 490, 14, 240, 17, 9},
                        {240, 490, 14, 480, 9, 5},
                        {480, 945, 27, 312, 5, 3}};
    float* bufs[2] = {P0, P1};

    for (int i = 0; i < 4; ++i) {
        const Blk& L = blk[i];
        float* bin  = bufs[i & 1];
        float* bout = bufs[(i + 1) & 1];
        const int nvox = L.inD * L.inD * L.inD;

        conv(3, bin, bw1[i], MID, L.Cin, L.Cmid, L.inD, L.inD, 1, 1);
        bn_stats<<<L.f, 256, 0, stream>>>(MID, STAT, L.f, nvox);
        bn_apply<<<4 * L.Cmid, 256, 0, stream>>>(MID, STAT, bnw[i], bnb[i],
                                                 L.Cmid, nvox);
        fourier(nvox, L.f, bA[i], bAi[i]);

        conv(3, MID, bw2[i], bout, L.Cmid, L.Cout, L.inD, L.outD, 1, 2);
        conv(1, bin, bsw[i], SB, L.Cin, L.Cout, L.inD, L.inD, 0, 1);
        avgpool2_add<<<4 * L.Cout, 256, 0, stream>>>(SB, bout, L.inD, L.outD);
    }

    // head: [4,312,3^3] -> [4,4480]
    conv(3, P0, w5, W5O, 312, 4480, 3, 1, 0, 1);
    norm_pool_k<<<2, 256, 0, stream>>>(W5O, FCH);
    fc_head<<<1, 128, 0, stream>>>(FCH, fc1w, fc1b, fc1gw, fc1gb,
                                   fc2w, fc2b, fc2gw, fc2gb, fc3w, fc3b,
                                   (float*)d_out);
}